// DMTetGeometry_970662609242
// MI455X (gfx1250) — compile-verified
//
#include <hip/hip_runtime.h>

// ---------------------------------------------------------------------------
// DMTet geometry extraction for MI455X (gfx1250).
// Memory-bound pipeline (~0.6 GB traffic -> ~25us floor at 23.3 TB/s); no
// dense matmul so no WMMA. CDNA5 paths: double-buffered TDM tensor_load_to_lds
// streaming of the tet array into LDS, s_wait_tensorcnt, global_prefetch_b8,
// wave32 block scans.
// ---------------------------------------------------------------------------

typedef unsigned int u32;
typedef __attribute__((ext_vector_type(4))) u32 u32x4;
typedef __attribute__((ext_vector_type(8))) int i32x8;
typedef __attribute__((ext_vector_type(4))) int i32x4;

#if __has_builtin(__builtin_amdgcn_tensor_load_to_lds) && __has_builtin(__builtin_amdgcn_s_wait_tensorcnt)
#define HAVE_TDM 1
#if __clang_major__ >= 23
#pragma message("CDNA5: TDM enabled, 6-arg tensor_load_to_lds (clang>=23)")
#else
#pragma message("CDNA5: TDM enabled, 5-arg tensor_load_to_lds (clang<=22)")
#endif
#else
#define HAVE_TDM 0
#warning "gfx1250 TDM builtins not found; falling back to plain LDS staging"
#endif

// ----- constants ------------------------------------------------------------
#define NVERTS 262144
#define NTETS  1300000

// header slots (u32) in workspace
#define H_M   0
#define H_N1  1
#define H_N2  2
#define H_C1  3
#define H_C3  4
#define H_NIN 5
#define H_U   6

__constant__ int d_NTRI[16] = {0,1,1,2,1,2,2,1,1,2,2,1,2,1,1,0};
__constant__ int d_NTET[16] = {0,1,1,3,1,3,3,3,1,3,3,3,3,3,3,1};
__constant__ int d_EP[12]   = {0,1, 0,2, 0,3, 1,2, 1,3, 2,3};

__constant__ int d_TRI[16*6] = {
  -1,-1,-1,-1,-1,-1,
   1, 0, 2,-1,-1,-1,
   4, 0, 3,-1,-1,-1,
   1, 4, 2, 1, 3, 4,
   3, 1, 5,-1,-1,-1,
   2, 3, 0, 2, 5, 3,
   1, 4, 0, 1, 5, 4,
   4, 2, 5,-1,-1,-1,
   4, 5, 2,-1,-1,-1,
   4, 1, 0, 4, 5, 1,
   3, 2, 0, 3, 5, 2,
   1, 3, 5,-1,-1,-1,
   4, 1, 2, 4, 3, 1,
   3, 0, 4,-1,-1,-1,
   2, 0, 1,-1,-1,-1,
  -1,-1,-1,-1,-1,-1};

__constant__ int d_TET[16*12] = {
  -1,-1,-1,-1,-1,-1,-1,-1,-1,-1,-1,-1,
   0, 4, 5, 6,-1,-1,-1,-1,-1,-1,-1,-1,
   1, 4, 8, 7,-1,-1,-1,-1,-1,-1,-1,-1,
   7, 1, 8, 6, 5, 1, 7, 6, 5, 0, 1, 6,
   2, 5, 7, 9,-1,-1,-1,-1,-1,-1,-1,-1,
   4, 0, 6, 7, 9, 0, 7, 6, 7, 0, 9, 2,
   4, 1, 9, 8, 5, 1, 9, 4, 5, 1, 2, 9,
   6, 0, 1, 2, 8, 6, 1, 2, 9, 6, 8, 2,
   3, 6, 9, 8,-1,-1,-1,-1,-1,-1,-1,-1,
   5, 0, 4, 8, 5, 0, 8, 3, 5, 8, 9, 3,
   1, 4, 7, 3, 4, 7, 6, 3, 9, 6, 7, 3,
   0, 1, 5, 3, 5, 1, 9, 3, 5, 1, 7, 9,
   5, 2, 3, 7, 3, 6, 5, 8, 3, 5, 7, 8,
   0, 4, 7, 8, 0, 3, 8, 7, 0, 3, 7, 2,
   4, 1, 2, 3, 4, 3, 2, 5, 4, 3, 5, 6,
   0, 1, 2, 3,-1,-1,-1,-1,-1,-1,-1,-1};

// ----- utility kernels ------------------------------------------------------
__global__ void zero_kernel(u32* __restrict__ p, long long L) {
  long long i = (long long)blockIdx.x * blockDim.x + threadIdx.x;
  if (i < L) p[i] = 0u;
}

__global__ void blend_sdf_kernel(const float* __restrict__ s1, const float* __restrict__ s2,
                                 const float* __restrict__ coef, float* __restrict__ sdf, int N) {
  int i = blockIdx.x * blockDim.x + threadIdx.x;
  if (i < N) {
    float c = coef[0];
    sdf[i] = c * s1[i] + (1.0f - c) * s2[i];
  }
}

// ----- exclusive scan (1024-elem tiles, 256 threads x 4) --------------------
__global__ __launch_bounds__(256) void scan_block_kernel(const u32* __restrict__ src,
                                                         u32* __restrict__ dst,
                                                         u32* __restrict__ blockSums, int L) {
  __shared__ u32 sh[256];
  int tid = threadIdx.x;
  int base = blockIdx.x * 1024 + tid * 4;
  u32 v[4];
  for (int k = 0; k < 4; ++k) { int i = base + k; v[k] = (i < L) ? src[i] : 0u; }
  u32 local = v[0] + v[1] + v[2] + v[3];
  sh[tid] = local;
  __syncthreads();
  for (int off = 1; off < 256; off <<= 1) {
    u32 t = (tid >= off) ? sh[tid - off] : 0u;
    __syncthreads();
    sh[tid] += t;
    __syncthreads();
  }
  u32 run = sh[tid] - local;
  for (int k = 0; k < 4; ++k) { int i = base + k; if (i < L) dst[i] = run; run += v[k]; }
  if (tid == 255) blockSums[blockIdx.x] = sh[255];
}

__global__ __launch_bounds__(256) void add_offsets_kernel(u32* __restrict__ dst,
                                                          const u32* __restrict__ sums, int L) {
  int base = blockIdx.x * 1024 + threadIdx.x * 4;
  u32 o = sums[blockIdx.x];
  for (int k = 0; k < 4; ++k) { int i = base + k; if (i < L) dst[i] += o; }
}

__global__ void store_total_kernel(u32* __restrict__ hdr, int slot,
                                   const u32* __restrict__ src, const u32* __restrict__ scanned, int L) {
  if (threadIdx.x == 0 && blockIdx.x == 0) hdr[slot] = scanned[L - 1] + src[L - 1];
}

// ----- TDM issue helper ------------------------------------------------------
#if HAVE_TDM
__device__ __forceinline__ void tdm_issue_tets(const long long* tet, int start, int n,
                                               long long* ldsDst) {
  unsigned ldsOff = (unsigned)(unsigned long long)ldsDst;
  unsigned long long ga = (unsigned long long)(const void*)tet + (unsigned long long)start * 32ull;
  u32x4 g0 = { 1u,                                   // count=1 (valid descriptor)
               ldsOff,                               // lds_addr (bytes)
               (u32)ga,                              // global_addr[31:0]
               ((u32)(ga >> 32) & 0x01FFFFFFu) | (2u << 30) };  // addr[56:32] | type=2
  i32x8 g1;
  g1[0] = 0x00030000;                                // data_size=3 (8B), no multicast/pad
  g1[1] = (4 << 16);                                 // tensor_dim0 = 4 (bits 79:48 low half)
  g1[2] = (int)(((u32)n & 0xFFFFu) << 16);           // tensor_dim0 hi=0 | tensor_dim1 lo
  g1[3] = (int)((((u32)n >> 16) & 0xFFFFu) | (4u << 16)); // tensor_dim1 hi | tile_dim0=4
  g1[4] = (int)((u32)n & 0xFFFFu);                   // tile_dim1 = tets in tile, tile_dim2=0
  g1[5] = 4;                                         // tensor_dim0_stride = 4 elements
  g1[6] = 0;
  g1[7] = 0;
  i32x4 z4 = {0, 0, 0, 0};
#if __clang_major__ >= 23
  i32x8 z8 = {0, 0, 0, 0, 0, 0, 0, 0};
  __builtin_amdgcn_tensor_load_to_lds(g0, g1, z4, z4, z8, 0);
#else
  __builtin_amdgcn_tensor_load_to_lds(g0, g1, z4, z4, 0);
#endif
}
#endif

// ----- tet classification (double-buffered TDM tiles of int64 tets) ---------
__device__ __forceinline__ void classify_one(const long long* __restrict__ row, int t,
                                             const float* __restrict__ sdf,
                                             u32* __restrict__ tetidx,
                                             u32* __restrict__ bucketCount) {
  int v[4];
  u32 occ = 0u;
  for (int j = 0; j < 4; ++j) {
    v[j] = (int)row[j];
    occ |= (sdf[v[j]] > 0.0f) ? (1u << j) : 0u;
  }
  tetidx[t] = occ;
  if (occ != 0u && occ != 15u) {
    for (int e = 0; e < 6; ++e) {
      int i0 = d_EP[2 * e], i1 = d_EP[2 * e + 1];
      if (((occ >> i0) ^ (occ >> i1)) & 1u) {
        int a = v[i0] < v[i1] ? v[i0] : v[i1];
        atomicAdd(&bucketCount[a], 1u);
      }
    }
  }
}

__global__ __launch_bounds__(256) void classify_kernel(const long long* __restrict__ tet,
                                                       const float* __restrict__ sdf,
                                                       u32* __restrict__ tetidx,
                                                       u32* __restrict__ bucketCount, int T) {
  __shared__ long long tile[2][256 * 4];
  int base0 = blockIdx.x * 512;
  int n0 = T - base0;         if (n0 > 256) n0 = 256;
  int n1 = T - (base0 + 256); if (n1 > 256) n1 = 256;
  if (n0 <= 0) return;  // block-uniform

#if HAVE_TDM
  bool wave0 = (threadIdx.x >> 5) == 0;
  if (wave0) {
    tdm_issue_tets(tet, base0, n0, tile[0]);          // DMA A
    if (n1 > 0) {
      tdm_issue_tets(tet, base0 + 256, n1, tile[1]);  // DMA B (overlaps compute of A)
      __builtin_amdgcn_s_wait_tensorcnt(1);           // A complete (in-order per wave)
    } else {
      __builtin_amdgcn_s_wait_tensorcnt(0);
    }
  }
  __syncthreads();
#else
  for (int i = threadIdx.x; i < n0 * 4; i += 256)
    tile[0][i] = tet[(long long)base0 * 4 + i];
  if (n1 > 0)
    for (int i = threadIdx.x; i < n1 * 4; i += 256)
      tile[1][i] = tet[(long long)(base0 + 256) * 4 + i];
  __syncthreads();
#endif

  if ((int)threadIdx.x < n0)
    classify_one(&tile[0][threadIdx.x * 4], base0 + threadIdx.x, sdf, tetidx, bucketCount);

  if (n1 > 0) {  // block-uniform
#if HAVE_TDM
    if (wave0) __builtin_amdgcn_s_wait_tensorcnt(0);  // B complete
#endif
    __syncthreads();
    if ((int)threadIdx.x < n1)
      classify_one(&tile[1][threadIdx.x * 4], base0 + 256 + threadIdx.x, sdf, tetidx, bucketCount);
  }
}

// ----- scatter edge instances into buckets ----------------------------------
__global__ __launch_bounds__(256) void scatter_edges_kernel(const long long* __restrict__ tet,
                                     const u32* __restrict__ tetidx,
                                     const u32* __restrict__ bucketStart, u32* __restrict__ cursor,
                                     u32* __restrict__ edgeB, int T) {
  int t = blockIdx.x * blockDim.x + threadIdx.x;
  if (t >= T) return;
  __builtin_prefetch((const void*)(tet + ((long long)t + 4096) * 4), 0, 1);
  u32 occ = tetidx[t];
  if (occ == 0u || occ == 15u) return;
  longlong4 q = ((const longlong4*)tet)[t];
  int v[4] = {(int)q.x, (int)q.y, (int)q.z, (int)q.w};
  for (int e = 0; e < 6; ++e) {
    int i0 = d_EP[2 * e], i1 = d_EP[2 * e + 1];
    if (((occ >> i0) ^ (occ >> i1)) & 1u) {
      int a = v[i0] < v[i1] ? v[i0] : v[i1];
      int b = v[i0] < v[i1] ? v[i1] : v[i0];
      u32 p = bucketStart[a] + atomicAdd(&cursor[a], 1u);
      edgeB[p] = (u32)b;
    }
  }
}

// ----- per-bucket sort + dedup (small buckets; one thread each) -------------
__global__ __launch_bounds__(256) void dedup_kernel(u32* __restrict__ edgeB,
                             const u32* __restrict__ bucketStart,
                             const u32* __restrict__ bucketCount, u32* __restrict__ uniqueCount, int N) {
  int a = blockIdx.x * blockDim.x + threadIdx.x;
  if (a >= N) return;
  u32 s = bucketStart[a];
  int c = (int)bucketCount[a];
  for (int i = 1; i < c; ++i) {          // insertion sort
    u32 key = edgeB[s + i];
    int j = i - 1;
    while (j >= 0 && edgeB[s + j] > key) { edgeB[s + j + 1] = edgeB[s + j]; --j; }
    edgeB[s + j + 1] = key;
  }
  u32 u = 0, prev = 0xFFFFFFFFu;
  for (int i = 0; i < c; ++i) {          // dedup in place
    u32 x = edgeB[s + i];
    if (x != prev) { edgeB[s + u] = x; ++u; prev = x; }
  }
  uniqueCount[a] = u;
}

// ----- interpolated vertices -------------------------------------------------
__global__ __launch_bounds__(256) void emit_verts_kernel(const float* __restrict__ pos,
                                  const float* __restrict__ sdf,
                                  const u32* __restrict__ bucketStart, const u32* __restrict__ uniqueCount,
                                  const u32* __restrict__ uniqueStart, const u32* __restrict__ edgeB,
                                  float* __restrict__ out, int N) {
  int a = blockIdx.x * blockDim.x + threadIdx.x;
  if (a >= N) return;
  u32 s = bucketStart[a], uc = uniqueCount[a], r0 = uniqueStart[a];
  float fa = sdf[a];
  for (u32 k = 0; k < uc; ++k) {
    int b = (int)edgeB[s + k];
    float fb = sdf[b];
    float denom = fa - fb;               // crossing edge -> nonzero
    float w0 = (-fb) / denom;
    float w1 = fa / denom;
    long long r = (long long)(r0 + k) * 3;
    for (int d = 0; d < 3; ++d)
      out[r + d] = pos[(long long)a * 3 + d] * w0 + pos[(long long)b * 3 + d] * w1;
  }
}

__device__ __forceinline__ int edge_rank(int a, int b, const u32* bucketStart,
                                         const u32* uniqueCount, const u32* uniqueStart,
                                         const u32* edgeB) {
  u32 s = bucketStart[a];
  u32 lo = 0, hi = uniqueCount[a];
  while (lo < hi) {
    u32 mid = (lo + hi) >> 1;
    if (edgeB[s + mid] < (u32)b) lo = mid + 1; else hi = mid;
  }
  return (int)(uniqueStart[a] + lo);
}

// ----- stable flags for compaction ------------------------------------------
__global__ __launch_bounds__(256) void make_flags_kernel(const u32* __restrict__ tetidx,
                                                         u32* __restrict__ flag, int T, int mode) {
  int t = blockIdx.x * blockDim.x + threadIdx.x;
  if (t >= T) return;
  u32 occ = tetidx[t];
  bool valid = (occ != 0u && occ != 15u);
  u32 f = 0u;
  switch (mode) {
    case 0: f = (valid && d_NTRI[occ] == 1) ? 1u : 0u; break;
    case 1: f = (valid && d_NTRI[occ] == 2) ? 1u : 0u; break;
    case 2: f = (valid && d_NTET[occ] == 1) ? 1u : 0u; break;
    case 3: f = (valid && d_NTET[occ] == 3) ? 1u : 0u; break;
    default: f = (occ == 15u) ? 1u : 0u; break;
  }
  flag[t] = f;
}

// ----- faces -----------------------------------------------------------------
__global__ __launch_bounds__(256) void emit_faces_kernel(const long long* __restrict__ tet,
                                  const u32* __restrict__ tetidx,
                                  const u32* __restrict__ bucketStart, const u32* __restrict__ uniqueCount,
                                  const u32* __restrict__ uniqueStart, const u32* __restrict__ edgeB,
                                  const u32* __restrict__ scan1, const u32* __restrict__ scan2,
                                  const u32* __restrict__ hdr, float* __restrict__ out, int T) {
  int t = blockIdx.x * blockDim.x + threadIdx.x;
  if (t >= T) return;
  u32 occ = tetidx[t];
  if (occ == 0u || occ == 15u) return;
  __builtin_prefetch((const void*)(tet + ((long long)t + 4096) * 4), 0, 1);
  longlong4 q = ((const longlong4*)tet)[t];
  int v[4] = {(int)q.x, (int)q.y, (int)q.z, (int)q.w};
  int r[6];
  for (int e = 0; e < 6; ++e) {
    int i0 = d_EP[2 * e], i1 = d_EP[2 * e + 1];
    if (((occ >> i0) ^ (occ >> i1)) & 1u) {
      int a = v[i0] < v[i1] ? v[i0] : v[i1];
      int b = v[i0] < v[i1] ? v[i1] : v[i0];
      r[e] = edge_rank(a, b, bucketStart, uniqueCount, uniqueStart, edgeB);
    } else r[e] = -1;
  }
  long long baseF = (long long)3 * (long long)hdr[H_M];
  int nt = d_NTRI[occ];
  if (nt == 1) {
    long long slot = (long long)scan1[t];
    for (int j = 0; j < 3; ++j)
      out[baseF + slot * 3 + j] = (float)r[d_TRI[occ * 6 + j]];
  } else {
    long long slot = (long long)hdr[H_N1] + (long long)scan2[t] * 2;
    for (int j = 0; j < 6; ++j)
      out[baseF + slot * 3 + j] = (float)r[d_TRI[occ * 6 + j]];
  }
}

// ----- side tets + inner tets -> allTets, mark used vertex ids --------------
__global__ __launch_bounds__(256) void emit_tets_kernel(const long long* __restrict__ tet,
                                 const u32* __restrict__ tetidx,
                                 const u32* __restrict__ bucketStart, const u32* __restrict__ uniqueCount,
                                 const u32* __restrict__ uniqueStart, const u32* __restrict__ edgeB,
                                 const u32* __restrict__ sc1, const u32* __restrict__ sc3,
                                 const u32* __restrict__ scin, const u32* __restrict__ hdr,
                                 int* __restrict__ allTets, u32* __restrict__ usedFlag, int T, int Nv) {
  int t = blockIdx.x * blockDim.x + threadIdx.x;
  if (t >= T) return;
  u32 occ = tetidx[t];
  if (occ == 0u) return;
  longlong4 q = ((const longlong4*)tet)[t];
  int v[4] = {(int)q.x, (int)q.y, (int)q.z, (int)q.w};
  if (occ == 15u) {  // fully inside tets come after all side tets
    long long slot = (long long)hdr[H_C1] + 3ll * (long long)hdr[H_C3] + (long long)scin[t];
    for (int j = 0; j < 4; ++j) { allTets[slot * 4 + j] = v[j]; usedFlag[v[j]] = 1u; }
    return;
  }
  int tve[10];
  for (int j = 0; j < 4; ++j) tve[j] = v[j];
  for (int e = 0; e < 6; ++e) {
    int i0 = d_EP[2 * e], i1 = d_EP[2 * e + 1];
    if (((occ >> i0) ^ (occ >> i1)) & 1u) {
      int a = v[i0] < v[i1] ? v[i0] : v[i1];
      int b = v[i0] < v[i1] ? v[i1] : v[i0];
      tve[4 + e] = Nv + edge_rank(a, b, bucketStart, uniqueCount, uniqueStart, edgeB);
    } else tve[4 + e] = 0;  // never referenced by the table for this occ
  }
  int nt = d_NTET[occ];
  if (nt == 1) {
    long long slot = (long long)sc1[t];
    for (int j = 0; j < 4; ++j) {
      int id = tve[d_TET[occ * 12 + j]];
      allTets[slot * 4 + j] = id; usedFlag[id] = 1u;
    }
  } else {
    long long base = ((long long)hdr[H_C1] + (long long)sc3[t] * 3) * 4;
    for (int j = 0; j < 12; ++j) {
      int id = tve[d_TET[occ * 12 + j]];
      allTets[base + j] = id; usedFlag[id] = 1u;
    }
  }
}

// ----- compacted tet-mesh vertices and tets ---------------------------------
__global__ __launch_bounds__(256) void emit_all_verts_kernel(const float* __restrict__ pos,
                                      const u32* __restrict__ usedFlag,
                                      const u32* __restrict__ usedRank, const u32* __restrict__ hdr,
                                      float* __restrict__ out, int Nv, long long L) {
  long long i = (long long)blockIdx.x * blockDim.x + threadIdx.x;
  if (i >= L || !usedFlag[i]) return;
  long long M = hdr[H_M];
  long long F = (long long)hdr[H_N1] + 2ll * (long long)hdr[H_N2];
  long long base = 3ll * M + 3ll * F;
  long long r = usedRank[i];
  if (i < Nv) {
    for (int d = 0; d < 3; ++d) out[base + r * 3 + d] = pos[i * 3 + d];
  } else {
    for (int d = 0; d < 3; ++d) out[base + r * 3 + d] = out[(i - Nv) * 3 + d];
  }
}

__global__ __launch_bounds__(256) void emit_all_tets_kernel(const int* __restrict__ allTets,
                                     const u32* __restrict__ usedRank,
                                     const u32* __restrict__ hdr, float* __restrict__ out, long long Kmax) {
  long long k = (long long)blockIdx.x * blockDim.x + threadIdx.x;
  long long K = (long long)hdr[H_C1] + 3ll * (long long)hdr[H_C3] + (long long)hdr[H_NIN];
  if (k >= K || k >= Kmax) return;
  long long M = hdr[H_M];
  long long F = (long long)hdr[H_N1] + 2ll * (long long)hdr[H_N2];
  long long U = hdr[H_U];
  long long base = 3ll * M + 3ll * F + 3ll * U;
  for (int j = 0; j < 4; ++j)
    out[base + k * 4 + j] = (float)usedRank[allTets[k * 4 + j]];
}

// ----- host-side multi-level exclusive scan ---------------------------------
static void scan_exclusive(const u32* src, u32* dst, u32* tmp, int L, hipStream_t st) {
  int nb = (L + 1023) / 1024;
  scan_block_kernel<<<nb, 256, 0, st>>>(src, dst, tmp, L);
  if (nb > 1) {
    u32* sums = tmp;
    u32* sumsScanned = tmp + nb;
    scan_exclusive(sums, sumsScanned, tmp + 2 * nb, nb, st);
    add_offsets_kernel<<<nb, 256, 0, st>>>(dst, sumsScanned, L);
  }
}

// ---------------------------------------------------------------------------
extern "C" void kernel_launch(void* const* d_in, const int* in_sizes, int n_in,
                              void* d_out, int out_size, void* d_ws, size_t ws_size,
                              hipStream_t stream) {
  (void)in_sizes; (void)n_in; (void)out_size; (void)ws_size;

  const float*     pos  = (const float*)d_in[0];
  const float*     s1   = (const float*)d_in[1];
  const float*     s2   = (const float*)d_in[2];
  const float*     coef = (const float*)d_in[3];
  const long long* tet  = (const long long*)d_in[4];  // int64 per reference dtype
  float* out = (float*)d_out;
  u32*   W   = (u32*)d_ws;

  constexpr int       NV    = NVERTS;
  constexpr int       NT    = NTETS;
  constexpr long long EIMAX = 6ll * NT;        // 7.8M edge instances
  constexpr long long KMAX  = 3ll * NT;        // max side+inner tets
  constexpr long long LUSED = (long long)NV + EIMAX;

  // workspace layout (u32 units)
  constexpr size_t OFF_HDR    = 0;
  constexpr size_t OFF_SDF    = 64;
  constexpr size_t OFF_TIDX   = OFF_SDF + NV;
  constexpr size_t OFF_SCANA  = OFF_TIDX + NT;
  constexpr size_t OFF_SCANB  = OFF_SCANA + NT;
  constexpr size_t OFF_SCANC  = OFF_SCANB + NT;
  constexpr size_t OFF_FLAG   = OFF_SCANC + NT;
  constexpr size_t OFF_BCOUNT = OFF_FLAG + NT;
  constexpr size_t OFF_BSTART = OFF_BCOUNT + NV;
  constexpr size_t OFF_BCUR   = OFF_BSTART + NV;
  constexpr size_t OFF_UCOUNT = OFF_BCUR + NV;
  constexpr size_t OFF_USTART = OFF_UCOUNT + NV;
  constexpr size_t OFF_EDGEB  = OFF_USTART + NV;
  constexpr size_t OFF_ALLT   = OFF_EDGEB + (size_t)EIMAX;
  constexpr size_t OFF_USEDF  = OFF_ALLT + (size_t)(4 * KMAX);
  constexpr size_t OFF_USEDR  = OFF_USEDF + (size_t)LUSED;
  constexpr size_t OFF_TMP    = OFF_USEDR + (size_t)LUSED;

  u32*   hdr  = W + OFF_HDR;
  float* sdf  = (float*)(W + OFF_SDF);
  u32*   tmp  = W + OFF_TMP;

  const int NB_NV  = (NV + 255) / 256;
  const int NB_NT  = (NT + 255) / 256;
  const int NB_CLS = (NT + 511) / 512;   // classify: 2 tiles (512 tets) per block
  const int NB_LU  = (int)((LUSED + 255) / 256);
  const int NB_K   = (int)((KMAX + 255) / 256);

  // 0) zero what must be zero each call (d_ws is poisoned once, not re-poisoned)
  zero_kernel<<<1, 64, 0, stream>>>(hdr, 64);
  zero_kernel<<<NB_NV, 256, 0, stream>>>(W + OFF_BCOUNT, NV);
  zero_kernel<<<NB_NV, 256, 0, stream>>>(W + OFF_BCUR, NV);
  zero_kernel<<<NB_LU, 256, 0, stream>>>(W + OFF_USEDF, LUSED);

  // 1) sdf blend
  blend_sdf_kernel<<<NB_NV, 256, 0, stream>>>(s1, s2, coef, sdf, NV);

  // 2) classify tets (double-buffered TDM staging) + bucket counts
  classify_kernel<<<NB_CLS, 256, 0, stream>>>(tet, sdf, W + OFF_TIDX, W + OFF_BCOUNT, NT);

  // 3) bucket starts; scatter edge instances; sort+dedup per bucket; edge ranks
  scan_exclusive(W + OFF_BCOUNT, W + OFF_BSTART, tmp, NV, stream);
  scatter_edges_kernel<<<NB_NT, 256, 0, stream>>>(tet, W + OFF_TIDX, W + OFF_BSTART,
                                                  W + OFF_BCUR, W + OFF_EDGEB, NT);
  dedup_kernel<<<NB_NV, 256, 0, stream>>>(W + OFF_EDGEB, W + OFF_BSTART, W + OFF_BCOUNT,
                                          W + OFF_UCOUNT, NV);
  scan_exclusive(W + OFF_UCOUNT, W + OFF_USTART, tmp, NV, stream);
  store_total_kernel<<<1, 1, 0, stream>>>(hdr, H_M, W + OFF_UCOUNT, W + OFF_USTART, NV);

  // 4) interpolated surface vertices -> out[0 : 3M)
  emit_verts_kernel<<<NB_NV, 256, 0, stream>>>(pos, sdf, W + OFF_BSTART, W + OFF_UCOUNT,
                                               W + OFF_USTART, W + OFF_EDGEB, out, NV);

  // 5) faces (ntri==1 block first, then ntri==2 — stable like the reference)
  make_flags_kernel<<<NB_NT, 256, 0, stream>>>(W + OFF_TIDX, W + OFF_FLAG, NT, 0);
  scan_exclusive(W + OFF_FLAG, W + OFF_SCANA, tmp, NT, stream);
  store_total_kernel<<<1, 1, 0, stream>>>(hdr, H_N1, W + OFF_FLAG, W + OFF_SCANA, NT);
  make_flags_kernel<<<NB_NT, 256, 0, stream>>>(W + OFF_TIDX, W + OFF_FLAG, NT, 1);
  scan_exclusive(W + OFF_FLAG, W + OFF_SCANB, tmp, NT, stream);
  store_total_kernel<<<1, 1, 0, stream>>>(hdr, H_N2, W + OFF_FLAG, W + OFF_SCANB, NT);
  emit_faces_kernel<<<NB_NT, 256, 0, stream>>>(tet, W + OFF_TIDX, W + OFF_BSTART, W + OFF_UCOUNT,
                                               W + OFF_USTART, W + OFF_EDGEB,
                                               W + OFF_SCANA, W + OFF_SCANB, hdr, out, NT);

  // 6) side tets (ntet==1, ntet==3) then inner tets; mark used vertex ids
  make_flags_kernel<<<NB_NT, 256, 0, stream>>>(W + OFF_TIDX, W + OFF_FLAG, NT, 2);
  scan_exclusive(W + OFF_FLAG, W + OFF_SCANA, tmp, NT, stream);
  store_total_kernel<<<1, 1, 0, stream>>>(hdr, H_C1, W + OFF_FLAG, W + OFF_SCANA, NT);
  make_flags_kernel<<<NB_NT, 256, 0, stream>>>(W + OFF_TIDX, W + OFF_FLAG, NT, 3);
  scan_exclusive(W + OFF_FLAG, W + OFF_SCANB, tmp, NT, stream);
  store_total_kernel<<<1, 1, 0, stream>>>(hdr, H_C3, W + OFF_FLAG, W + OFF_SCANB, NT);
  make_flags_kernel<<<NB_NT, 256, 0, stream>>>(W + OFF_TIDX, W + OFF_FLAG, NT, 4);
  scan_exclusive(W + OFF_FLAG, W + OFF_SCANC, tmp, NT, stream);
  store_total_kernel<<<1, 1, 0, stream>>>(hdr, H_NIN, W + OFF_FLAG, W + OFF_SCANC, NT);
  emit_tets_kernel<<<NB_NT, 256, 0, stream>>>(tet, W + OFF_TIDX, W + OFF_BSTART, W + OFF_UCOUNT,
                                              W + OFF_USTART, W + OFF_EDGEB,
                                              W + OFF_SCANA, W + OFF_SCANB, W + OFF_SCANC, hdr,
                                              (int*)(W + OFF_ALLT), W + OFF_USEDF, NT, NV);

  // 7) compact vertex ids (jnp.unique over all_tets == sorted used-id scan)
  scan_exclusive(W + OFF_USEDF, W + OFF_USEDR, tmp, (int)LUSED, stream);
  store_total_kernel<<<1, 1, 0, stream>>>(hdr, H_U, W + OFF_USEDF, W + OFF_USEDR, (int)LUSED);
  emit_all_verts_kernel<<<NB_LU, 256, 0, stream>>>(pos, W + OFF_USEDF, W + OFF_USEDR, hdr,
                                                   out, NV, LUSED);
  emit_all_tets_kernel<<<NB_K, 256, 0, stream>>>((const int*)(W + OFF_ALLT), W + OFF_USEDR,
                                                 hdr, out, KMAX);
}